// PointPatchEmbed_48077863911649
// MI455X (gfx1250) — compile-verified
//
#include <hip/hip_runtime.h>
#include <math.h>

// Problem constants
#define BB 32
#define PP 8192
#define MM 64
#define KK 32
#define CFEAT 13
#define CIN 16
#define EE 384
#define EPSV 1e-5f

typedef __attribute__((ext_vector_type(2))) float v2f;
typedef __attribute__((ext_vector_type(8))) float v8f;

// ---------------------------------------------------------------------------
// Kernel 1: fold BN into linear layers, store W'^T (Cin x Cout, row-major)
// o in [0,64): stage1; [64,192): stage2; [192,576): stage3
// ---------------------------------------------------------------------------
__global__ void prep_kernel(const float* __restrict__ W1, const float* __restrict__ b1,
                            const float* __restrict__ g1, const float* __restrict__ be1,
                            const float* __restrict__ m1, const float* __restrict__ v1,
                            const float* __restrict__ W2, const float* __restrict__ b2,
                            const float* __restrict__ g2, const float* __restrict__ be2,
                            const float* __restrict__ m2, const float* __restrict__ v2,
                            const float* __restrict__ W3, const float* __restrict__ b3,
                            const float* __restrict__ g3, const float* __restrict__ be3,
                            const float* __restrict__ m3, const float* __restrict__ v3,
                            float* __restrict__ WT1, float* __restrict__ B1,
                            float* __restrict__ WT2, float* __restrict__ B2,
                            float* __restrict__ WT3, float* __restrict__ B3) {
    int o = blockIdx.x * blockDim.x + threadIdx.x;
    if (o < 64) {
        float s = g1[o] / sqrtf(v1[o] + EPSV);
        B1[o] = s * (b1[o] - m1[o]) + be1[o];
        for (int c = 0; c < 16; ++c) WT1[c * 64 + o] = s * W1[o * 16 + c];
    } else if (o < 192) {
        int oo = o - 64;
        float s = g2[oo] / sqrtf(v2[oo] + EPSV);
        B2[oo] = s * (b2[oo] - m2[oo]) + be2[oo];
        for (int c = 0; c < 64; ++c) WT2[c * 128 + oo] = s * W2[oo * 64 + c];
    } else if (o < 576) {
        int oo = o - 192;
        float s = g3[oo] / sqrtf(v3[oo] + EPSV);
        B3[oo] = s * (b3[oo] - m3[oo]) + be3[oo];
        for (int c = 0; c < 128; ++c) WT3[c * 384 + oo] = s * W3[oo * 128 + c];
    }
}

// ---------------------------------------------------------------------------
// Kernel 2: centers output (tail of d_out)
// ---------------------------------------------------------------------------
__global__ void centers_kernel(const float* __restrict__ xyz, float* __restrict__ out) {
    int i = blockIdx.x * blockDim.x + threadIdx.x;
    if (i >= BB * MM * 3) return;
    int c = i % 3;
    int bm = i / 3;
    int m = bm % MM;
    int b = bm / MM;
    out[(size_t)BB * MM * EE + i] = xyz[((size_t)b * PP + (size_t)m * (PP / MM)) * 3 + c];
}

// ---------------------------------------------------------------------------
// Kernel 3: per-(b,m) exact top-K=32 nearest neighbors + gather of 16-ch rows
// d2 = |c|^2 + |p|^2 - 2 c.p  (same formula as reference)
// tie-break: smaller index wins (matches jax.lax.top_k stability)
// ---------------------------------------------------------------------------
__global__ void topk_gather_kernel(const float* __restrict__ xyz,
                                   const float* __restrict__ feat,
                                   float* __restrict__ xbuf) {
    __shared__ float d2s[PP];
    __shared__ float rv[256];
    __shared__ int   ri[256];
    __shared__ int   sel[KK];
    __shared__ float ctr[3];

    const int g   = blockIdx.x;      // b*M + m
    const int b   = g >> 6;
    const int m   = g & 63;
    const int tid = threadIdx.x;

    const float* pts = xyz + (size_t)b * PP * 3;
    if (tid < 3) ctr[tid] = pts[(size_t)m * (PP / MM) * 3 + tid];
    __syncthreads();

    const float cx = ctr[0], cy = ctr[1], cz = ctr[2];
    const float cc = cx * cx + cy * cy + cz * cz;

    for (int p = tid; p < PP; p += 256) {
        float px = pts[p * 3 + 0], py = pts[p * 3 + 1], pz = pts[p * 3 + 2];
        float pp = px * px + py * py + pz * pz;
        float dt = cx * px + cy * py + cz * pz;
        d2s[p] = cc + pp - 2.0f * dt;
    }
    __syncthreads();

    const float INF = __builtin_inff();
    for (int r = 0; r < KK; ++r) {
        float bv = INF; int bi = PP;
        for (int p = tid; p < PP; p += 256) {
            float v = d2s[p];
            if (v < bv) { bv = v; bi = p; }   // strided scan: index increases, ties keep first
        }
        rv[tid] = bv; ri[tid] = bi;
        __syncthreads();
        for (int s = 128; s > 0; s >>= 1) {
            if (tid < s) {
                float v2 = rv[tid + s]; int i2 = ri[tid + s];
                if (v2 < rv[tid] || (v2 == rv[tid] && i2 < ri[tid])) { rv[tid] = v2; ri[tid] = i2; }
            }
            __syncthreads();
        }
        if (tid == 0) { sel[r] = ri[0]; d2s[ri[0]] = INF; }
        __syncthreads();
    }

    // Build x rows: [nb_xyz - center (3) | nb_feat (13)] -> 16 channels
    const float* fpt = feat + (size_t)b * PP * CFEAT;
    for (int s = tid; s < KK * CIN; s += 256) {
        int j = s >> 4, c = s & 15;
        int p = sel[j];
        float val = (c < 3) ? (pts[p * 3 + c] - ctr[c]) : fpt[p * CFEAT + (c - 3)];
        xbuf[((size_t)g * KK + j) * CIN + c] = val;
    }
}

// ---------------------------------------------------------------------------
// Kernel 4: WMMA MLP + max-pool.  64 threads (2 waves) per (b,m) group.
// Wave w owns rows [w*16, w*16+16) of the 32-row group.
// f32 WMMA: D(16x16) = A(16x4) * B(4x16) + C, per ISA layouts:
//   A: lanes 0-15 -> M=lane, K = kb+{0,1}; lanes 16-31 -> M=lane-16, K = kb+{2,3}
//   B: VGPR v: lanes 0-15 -> K=kb+v, N=lane; lanes 16-31 -> K=kb+v+2, N=lane-16
//   C/D: VGPR r: lanes 0-15 -> M=r, N=lane; lanes 16-31 -> M=r+8, N=lane-16
// ---------------------------------------------------------------------------
template <int CI, int CO, int PIN, int POUT, bool GELU>
__device__ __forceinline__ void wmma_stage(const float* __restrict__ in,
                                           float* __restrict__ outp,
                                           const float* __restrict__ WT,
                                           const float* __restrict__ bias,
                                           int wave, int lane) {
    const int ln = lane & 15;
    const int hi = lane >> 4;            // 0 or 1
    const int r0 = wave * 16;
#pragma unroll
    for (int t = 0; t < CO / 16; ++t) {
        const int col = t * 16 + ln;
        const float bv = bias[col];
        v8f acc;
#pragma unroll
        for (int i = 0; i < 8; ++i) acc[i] = bv;
#pragma unroll
        for (int kb = 0; kb < CI; kb += 4) {
            const int k0 = kb + (hi ? 2 : 0);
            v2f a, bm;
            a.x  = in[(r0 + ln) * PIN + k0];
            a.y  = in[(r0 + ln) * PIN + k0 + 1];
            bm.x = WT[(size_t)k0 * CO + col];
            bm.y = WT[(size_t)(k0 + 1) * CO + col];
            acc = __builtin_amdgcn_wmma_f32_16x16x4_f32(false, a, false, bm,
                                                        (short)0, acc, false, false);
        }
#pragma unroll
        for (int r = 0; r < 8; ++r) {
            float v = acc[r];
            if (GELU) v = 0.5f * v * (1.0f + erff(v * 0.70710678118654752f));
            outp[(r0 + (hi ? 8 : 0) + r) * POUT + col] = v;
        }
    }
}

__global__ void mlp_kernel(const float* __restrict__ xbuf,
                           const float* __restrict__ WT1, const float* __restrict__ B1,
                           const float* __restrict__ WT2, const float* __restrict__ B2,
                           const float* __restrict__ WT3, const float* __restrict__ B3,
                           float* __restrict__ out) {
    __shared__ float Xs[32][17];    // 32 rows x 16 ch  (+1 pad)
    __shared__ float H1[32][65];    // 32 x 64  (+1 pad)
    __shared__ float H2[32][129];   // 32 x 128 (+1 pad)
    __shared__ float pmax[2][EE];   // per-wave partial max

    const int g    = blockIdx.x;           // b*M + m
    const int tid  = threadIdx.x;          // 0..63
    const int lane = tid & 31;
    const int wave = tid >> 5;             // 0 or 1
    const int ln   = lane & 15;
    const int hi   = lane >> 4;
    const int r0   = wave * 16;

    // Stage 0: load the 32x16 input tile
    for (int i = tid; i < 32 * 16; i += 64)
        Xs[i >> 4][i & 15] = xbuf[(size_t)g * (32 * 16) + i];
    __syncthreads();

    // Stage 1: 16 -> 64, GELU
    wmma_stage<16, 64, 17, 65, true>(&Xs[0][0], &H1[0][0], WT1, B1, wave, lane);
    __syncthreads();

    // Stage 2: 64 -> 128, GELU
    wmma_stage<64, 128, 65, 129, true>(&H1[0][0], &H2[0][0], WT2, B2, wave, lane);
    __syncthreads();

    // Stage 3: 128 -> 384, fused max over this wave's 16 rows
#pragma unroll 1
    for (int t = 0; t < EE / 16; ++t) {
        const int col = t * 16 + ln;
        const float bv = B3[col];
        v8f acc;
#pragma unroll
        for (int i = 0; i < 8; ++i) acc[i] = bv;
#pragma unroll 4
        for (int kb = 0; kb < 128; kb += 4) {
            const int k0 = kb + (hi ? 2 : 0);
            v2f a, bm;
            a.x  = H2[r0 + ln][k0];
            a.y  = H2[r0 + ln][k0 + 1];
            bm.x = WT3[(size_t)k0 * EE + col];
            bm.y = WT3[(size_t)(k0 + 1) * EE + col];
            acc = __builtin_amdgcn_wmma_f32_16x16x4_f32(false, a, false, bm,
                                                        (short)0, acc, false, false);
        }
        // per-lane max over 8 rows, then combine lanes L and L+16 (rows r / r+8)
        float mx = acc[0];
#pragma unroll
        for (int r = 1; r < 8; ++r) mx = fmaxf(mx, acc[r]);
        mx = fmaxf(mx, __shfl_xor(mx, 16, 32));
        if (hi == 0) pmax[wave][col] = mx;
    }
    __syncthreads();

    // Combine the two waves' partial maxes -> tokens
    for (int c = tid; c < EE; c += 64)
        out[(size_t)g * EE + c] = fmaxf(pmax[0][c], pmax[1][c]);
}

// ---------------------------------------------------------------------------
// Host-side launch
// ---------------------------------------------------------------------------
extern "C" void kernel_launch(void* const* d_in, const int* in_sizes, int n_in,
                              void* d_out, int out_size, void* d_ws, size_t ws_size,
                              hipStream_t stream) {
    const float* xyz  = (const float*)d_in[0];
    const float* feat = (const float*)d_in[1];
    // d_in[2] = batch (unused by reference math)
    const float* W1 = (const float*)d_in[3];  const float* b1 = (const float*)d_in[4];
    const float* g1 = (const float*)d_in[5];  const float* be1 = (const float*)d_in[6];
    const float* m1 = (const float*)d_in[7];  const float* v1 = (const float*)d_in[8];
    const float* W2 = (const float*)d_in[9];  const float* b2 = (const float*)d_in[10];
    const float* g2 = (const float*)d_in[11]; const float* be2 = (const float*)d_in[12];
    const float* m2 = (const float*)d_in[13]; const float* v2 = (const float*)d_in[14];
    const float* W3 = (const float*)d_in[15]; const float* b3 = (const float*)d_in[16];
    const float* g3 = (const float*)d_in[17]; const float* be3 = (const float*)d_in[18];
    const float* m3 = (const float*)d_in[19]; const float* v3 = (const float*)d_in[20];

    float* out = (float*)d_out;
    float* ws  = (float*)d_ws;

    // workspace layout (floats)
    float* xbuf = ws;                                  // 65536 * 16
    float* WT1  = xbuf + (size_t)BB * MM * KK * CIN;   // 16*64
    float* B1f  = WT1 + 16 * 64;                       // 64
    float* WT2  = B1f + 64;                            // 64*128
    float* B2f  = WT2 + 64 * 128;                      // 128
    float* WT3  = B2f + 128;                           // 128*384
    float* B3f  = WT3 + 128 * 384;                     // 384

    prep_kernel<<<9, 64, 0, stream>>>(W1, b1, g1, be1, m1, v1,
                                      W2, b2, g2, be2, m2, v2,
                                      W3, b3, g3, be3, m3, v3,
                                      WT1, B1f, WT2, B2f, WT3, B3f);

    centers_kernel<<<(BB * MM * 3 + 255) / 256, 256, 0, stream>>>(xyz, out);

    topk_gather_kernel<<<BB * MM, 256, 0, stream>>>(xyz, feat, xbuf);

    mlp_kernel<<<BB * MM, 64, 0, stream>>>(xbuf, WT1, B1f, WT2, B2f, WT3, B3f, out);
}